// SimpleBaseline_76639396429977
// MI455X (gfx1250) — compile-verified
//
#include <hip/hip_runtime.h>

typedef __attribute__((ext_vector_type(2))) float v2f;
typedef __attribute__((ext_vector_type(4))) float v4f;
typedef __attribute__((ext_vector_type(8))) float v8f;

#define NUM_G 2048
#define DIN   1536
#define DH1   64
#define DH2   32
#define ROWS_PER_BLOCK 128
#define SEG_THREADS 384   // 384 * 4 floats = 1536 columns

// ---------------------------------------------------------------------------
// Zero the workspace region (counts + sums).
// ---------------------------------------------------------------------------
__global__ void zero_ws_kernel(float* __restrict__ ws, int n) {
  int i = blockIdx.x * blockDim.x + threadIdx.x;
  if (i < n) ws[i] = 0.0f;
}

// ---------------------------------------------------------------------------
// Streaming segment-sum over sorted batch ids.
// Block = 384 threads; thread t owns columns [4t, 4t+4). 128 rows per block.
// Accumulate a float4 per thread for the current run of equal segment ids;
// flush with global float atomics only at segment boundaries.
// ---------------------------------------------------------------------------
__global__ void __launch_bounds__(SEG_THREADS)
segsum_kernel(const float* __restrict__ feat,
              const long long* __restrict__ batch,
              float* __restrict__ sums,
              float* __restrict__ counts,
              int nrows) {
  const int tid = threadIdx.x;
  int row = blockIdx.x * ROWS_PER_BLOCK;
  if (row >= nrows) return;
  int rend = row + ROWS_PER_BLOCK;
  if (rend > nrows) rend = nrows;

  const int col = tid * 4;
  v4f acc; acc.x = 0.f; acc.y = 0.f; acc.z = 0.f; acc.w = 0.f;
  int cur = (int)batch[row];
  int runStart = row;

  for (; row < rend; ++row) {
    const int s = (int)batch[row];
    if (s != cur) {
      float* dst = sums + (size_t)cur * DIN + col;
      atomicAdd(dst + 0, acc.x);
      atomicAdd(dst + 1, acc.y);
      atomicAdd(dst + 2, acc.z);
      atomicAdd(dst + 3, acc.w);
      if (tid == 0) atomicAdd(&counts[cur], (float)(row - runStart));
      acc.x = 0.f; acc.y = 0.f; acc.z = 0.f; acc.w = 0.f;
      cur = s;
      runStart = row;
    }
    if (row + 2 < rend)
      __builtin_prefetch(feat + (size_t)(row + 2) * DIN + col, 0, 0);
    const v4f v = *(const v4f*)(feat + (size_t)row * DIN + col);
    acc.x += v.x; acc.y += v.y; acc.z += v.z; acc.w += v.w;
  }
  // final flush
  float* dst = sums + (size_t)cur * DIN + col;
  atomicAdd(dst + 0, acc.x);
  atomicAdd(dst + 1, acc.y);
  atomicAdd(dst + 2, acc.z);
  atomicAdd(dst + 3, acc.w);
  if (tid == 0) atomicAdd(&counts[cur], (float)(rend - runStart));
}

// ---------------------------------------------------------------------------
// Tiny MLP over the 2048 pooled vectors, using fp32 WMMA (V_WMMA_F32_16X16X4_F32).
// Block = 128 threads (4 waves) handles a 16-graph M-tile.
//   Layer1: each wave owns one 16-wide N-tile of the 64 hidden units,
//           K-loop over 1536 in steps of 4. A = sums * (1/max(count,1))
//           (fused mean), B = W1^T columns. ReLU+bias -> LDS.
//   Layer2: waves 0,1 own the two 16-wide N-tiles of 32; K=64 from LDS.
//   Layer3: wave 0 lanes 0..15 do the 32-long dot with W3.
// ---------------------------------------------------------------------------
__global__ void __launch_bounds__(128)
mlp_kernel(const float* __restrict__ sums, const float* __restrict__ counts,
           const float* __restrict__ W1, const float* __restrict__ b1,
           const float* __restrict__ W2, const float* __restrict__ b2,
           const float* __restrict__ W3, const float* __restrict__ b3,
           float* __restrict__ out) {
  __shared__ float h1s[16 * DH1];
  __shared__ float h2s[16 * DH2];

  const int tid   = threadIdx.x;
  const int lane  = tid & 31;
  const int wid   = tid >> 5;           // 0..3, wave-uniform
  const int g0    = blockIdx.x * 16;    // first graph of this M-tile
  const int m     = lane & 15;          // A-fragment row / C column index
  const int hi    = lane >> 4;          // 0 for lanes 0-15, 1 for lanes 16-31
  const int khalf = hi * 2;             // K sub-offset within a 4-step (0 or 2)

  // fused mean: per-lane inverse count for row m
  const float inv = 1.0f / fmaxf(counts[g0 + m], 1.0f);
  const float* Arow = sums + (size_t)(g0 + m) * DIN;

  // ---------------- layer 1: [16 x 1536] x [1536 x 16] per wave ----------------
  {
    const int n = wid * 16 + m;                  // global hidden unit this lane feeds
    const float* Brow = W1 + (size_t)n * DIN;    // W1 is [64,1536] row-major
    v8f c = {};
    for (int k0 = 0; k0 < DIN; k0 += 4) {
      v2f a, b;
      a.x = Arow[k0 + khalf]     * inv;
      a.y = Arow[k0 + khalf + 1] * inv;
      b.x = Brow[k0 + khalf];
      b.y = Brow[k0 + khalf + 1];
      c = __builtin_amdgcn_wmma_f32_16x16x4_f32(false, a, false, b,
                                                (short)0, c, false, false);
    }
    const int nb = wid * 16;
#pragma unroll
    for (int r = 0; r < 8; ++r) {
      const int mr = r + hi * 8;                 // C/D row (graph within tile)
      const float v = c[r] + b1[nb + m];
      h1s[mr * DH1 + nb + m] = fmaxf(v, 0.0f);
    }
  }
  __syncthreads();

  // ---------------- layer 2: [16 x 64] x [64 x 16] on waves 0,1 ----------------
  if (wid < 2) {
    const int n = wid * 16 + m;
    const float* Brow = W2 + (size_t)n * DH1;    // W2 is [32,64] row-major
    v8f c = {};
    for (int k0 = 0; k0 < DH1; k0 += 4) {
      v2f a, b;
      a.x = h1s[m * DH1 + k0 + khalf];
      a.y = h1s[m * DH1 + k0 + khalf + 1];
      b.x = Brow[k0 + khalf];
      b.y = Brow[k0 + khalf + 1];
      c = __builtin_amdgcn_wmma_f32_16x16x4_f32(false, a, false, b,
                                                (short)0, c, false, false);
    }
#pragma unroll
    for (int r = 0; r < 8; ++r) {
      const int mr = r + hi * 8;
      const float v = c[r] + b2[wid * 16 + m];
      h2s[mr * DH2 + wid * 16 + m] = fmaxf(v, 0.0f);
    }
  }
  __syncthreads();

  // ---------------- layer 3: 32-long dot per graph ----------------
  if (wid == 0 && lane < 16) {
    float s = b3[0];
#pragma unroll
    for (int j = 0; j < DH2; ++j) s += h2s[lane * DH2 + j] * W3[j];
    out[g0 + lane] = s;
  }
}

// ---------------------------------------------------------------------------
extern "C" void kernel_launch(void* const* d_in, const int* in_sizes, int n_in,
                              void* d_out, int out_size, void* d_ws, size_t ws_size,
                              hipStream_t stream) {
  const float*     feat  = (const float*)d_in[0];
  const long long* batch = (const long long*)d_in[1];
  const float*     W1    = (const float*)d_in[2];
  const float*     b1    = (const float*)d_in[3];
  const float*     W2    = (const float*)d_in[4];
  const float*     b2    = (const float*)d_in[5];
  const float*     W3    = (const float*)d_in[6];
  const float*     b3    = (const float*)d_in[7];
  float*           out   = (float*)d_out;

  const int nrows = in_sizes[0] / DIN;

  // workspace layout: [counts: NUM_G floats][sums: NUM_G*DIN floats]
  float* counts = (float*)d_ws;
  float* sums   = counts + NUM_G;

  const int ztotal = NUM_G * DIN + NUM_G;
  zero_ws_kernel<<<(ztotal + 255) / 256, 256, 0, stream>>>((float*)d_ws, ztotal);

  const int nblocks = (nrows + ROWS_PER_BLOCK - 1) / ROWS_PER_BLOCK;
  segsum_kernel<<<nblocks, SEG_THREADS, 0, stream>>>(feat, batch, sums, counts, nrows);

  mlp_kernel<<<NUM_G / 16, 128, 0, stream>>>(sums, counts, W1, b1, W2, b2, W3, b3, out);
}